// BiRNNScratch_25529285608034
// MI455X (gfx1250) — compile-verified
//
#include <hip/hip_runtime.h>
#include <hip/hip_bf16.h>

// ---------------------------------------------------------------------------
// Bi-directional RNN scan, CDNA5 (gfx1250), wave32.
//   Phase 1 (xw_gemm): out[t,b, dir*512+n] = bias[n] + sum_k in[t,b,k]*W_xh[k,n]
//   Phase 2 (rnn_scan): sequential over t, 8 WGs per direction, each owning a
//     64-column slice. W_hh slice persisted (transposed) in LDS (CDNA5: 320KB
//     per WG), h tiles double-buffered through LDS via async-to-LDS,
//     per-step device-scope atomic barrier, WMMA f32 16x16x4.
// ---------------------------------------------------------------------------

#define TT    512
#define BB    64
#define NI    512
#define NH    512
#define OUTW  1024              // 2*NH
#define MROWS (TT*BB)           // 32768
#define NWG_DIR 8               // workgroups per direction in the scan
#define NSL   (NH/NWG_DIR)      // 64 columns per scan-WG

#define WT_STRIDE 516           // LDS stride for transposed W_hh (bank-safe)
#define HS_STRIDE 36            // LDS stride for h tiles (bank-safe)
#define HS_BUF    (64 * HS_STRIDE)
#define SCAN_LDS_FLOATS (64 * WT_STRIDE + 2 * HS_BUF)
#define SCAN_LDS_BYTES  (SCAN_LDS_FLOATS * 4)   // 150528 B < 320KB/WG

typedef float v2f __attribute__((ext_vector_type(2)));
typedef float v8f __attribute__((ext_vector_type(8)));
typedef int   v4i __attribute__((vector_size(4 * sizeof(int))));

// D = A(16x4 f32) * B(4x16 f32) + C(16x16 f32)  -- full f32 precision WMMA
static __device__ __forceinline__ v8f wmma4(v2f a, v2f b, v8f c) {
  return __builtin_amdgcn_wmma_f32_16x16x4_f32(
      /*neg_a=*/false, a, /*neg_b=*/false, b,
      /*c_mod=*/(short)0, c, /*reuse_a=*/false, /*reuse_b=*/false);
}

// ---- gfx1250 async global->LDS (GLOBAL_LOAD_ASYNC_TO_LDS_B128) -------------
// Prototype discovered via compile probe: arg0 = addrspace(1) v4i* (global),
// arg1 = LDS pointer, then i32 offset + i32 cpol.
#if defined(__HIP_DEVICE_COMPILE__) && \
    __has_builtin(__builtin_amdgcn_global_load_async_to_lds_b128)
#define HAVE_ASYNC 1
static __device__ __forceinline__ void async_cp16(const float* g, float* l) {
  __builtin_amdgcn_global_load_async_to_lds_b128(
      (__attribute__((address_space(1))) v4i*)(void*)g,
      (__attribute__((address_space(3))) v4i*)(void*)l,
      /*offset=*/0, /*cpol=*/0);
}
#else
#define HAVE_ASYNC 0
#endif

// ---------------------------------------------------------------------------
// Phase 0: zero the per-step arrival counters (d_ws is poisoned by harness)
// ---------------------------------------------------------------------------
__global__ void birnn_init_counters(unsigned* __restrict__ cnt) {
  const int i = threadIdx.x;
  if (i < 2 * TT) cnt[i] = 0u;
}

// ---------------------------------------------------------------------------
// Phase 1: xw GEMM.  Grid (256, 16): x = 128-row M tile, y = dir*8 + n-tile.
// Block tile 128(M) x 64(N), 8 waves, each wave a 32x32 patch (2x2 WMMA tiles).
// ---------------------------------------------------------------------------
__global__ __launch_bounds__(256) void birnn_xw_gemm(
    const float* __restrict__ X,      // (32768, 512)
    const float* __restrict__ Wf,     // (512, 512)
    const float* __restrict__ Wb,     // (512, 512)
    const float* __restrict__ bf,     // (512,)
    const float* __restrict__ bb,     // (512,)
    float* __restrict__ out)          // (32768, 1024) region of d_out
{
  __shared__ float As[128][36];       // stride 36: conflict-free f32 A-frags
  __shared__ float Bs[32][72];        // stride 72: half-waves hit disjoint banks

  const int tid  = threadIdx.x;
  const int wv   = tid >> 5;
  const int lane = tid & 31;
  const int l15  = lane & 15;
  const int hs   = lane >> 4;         // half-wave select

  const int m0  = blockIdx.x * 128;
  const int dir = blockIdx.y >> 3;
  const int n0  = (blockIdx.y & 7) * 64;       // column base within NH
  const float* __restrict__ W    = dir ? Wb : Wf;
  const float* __restrict__ bias = dir ? bb : bf;

  const int mw = (wv >> 1) * 32;      // 4 wave-rows
  const int nw = (wv & 1) * 32;       // 2 wave-cols

  v8f acc[2][2];
#pragma unroll
  for (int i = 0; i < 2; ++i)
#pragma unroll
    for (int j = 0; j < 2; ++j)
#pragma unroll
      for (int r = 0; r < 8; ++r) acc[i][j][r] = 0.0f;

  for (int kb = 0; kb < NI; kb += 32) {
    __syncthreads();
    // stage A: 128 x 32   (1024 float4, 4 per thread)
#pragma unroll
    for (int q = 0; q < 4; ++q) {
      const int idx = q * 256 + tid;
      const int r = idx >> 3, c4 = (idx & 7) * 4;
      const float4 v = *(const float4*)&X[(size_t)(m0 + r) * NI + kb + c4];
      *(float4*)&As[r][c4] = v;
    }
    // stage B: 32 x 64    (512 float4, 2 per thread)
#pragma unroll
    for (int q = 0; q < 2; ++q) {
      const int idx = q * 256 + tid;
      const int r = idx >> 4, c4 = (idx & 15) * 4;
      const float4 v = *(const float4*)&W[(size_t)(kb + r) * NH + n0 + c4];
      *(float4*)&Bs[r][c4] = v;
    }
    __syncthreads();

    // CDNA5 prefetch of next tiles (global_prefetch_b8)
    if (kb + 32 < NI) {
      __builtin_prefetch(&X[(size_t)(m0 + (tid >> 3)) * NI + kb + 32 + (tid & 7) * 4], 0, 0);
      __builtin_prefetch(&W[(size_t)(kb + 32 + (tid >> 4)) * NH + n0 + (tid & 15) * 4], 0, 0);
    }

#pragma unroll
    for (int kk = 0; kk < 8; ++kk) {
      const int kc = kk * 4 + 2 * hs;           // per-half K pair
      const v2f a0 = *(const v2f*)&As[mw + l15][kc];
      const v2f a1 = *(const v2f*)&As[mw + 16 + l15][kc];
      v2f b0, b1;
      b0.x = Bs[kc][nw + l15];       b0.y = Bs[kc + 1][nw + l15];
      b1.x = Bs[kc][nw + 16 + l15];  b1.y = Bs[kc + 1][nw + 16 + l15];
      acc[0][0] = wmma4(a0, b0, acc[0][0]);
      acc[0][1] = wmma4(a0, b1, acc[0][1]);
      acc[1][0] = wmma4(a1, b0, acc[1][0]);
      acc[1][1] = wmma4(a1, b1, acc[1][1]);
    }
  }

  // epilogue: += bias, store to out[:, dir*512 + n]
#pragma unroll
  for (int i = 0; i < 2; ++i)
#pragma unroll
    for (int j = 0; j < 2; ++j) {
      const int nc = n0 + nw + j * 16 + l15;    // within NH
      const float bv = bias[nc];
#pragma unroll
      for (int r = 0; r < 8; ++r) {
        const int mr = m0 + mw + i * 16 + hs * 8 + r;
        out[(size_t)mr * OUTW + dir * NH + nc] = acc[i][j][r] + bv;
      }
    }
}

// ---------------------------------------------------------------------------
// Phase 2: recurrent scan.  Grid 16 blocks: blockIdx.x = dir*8 + wg.
// Each WG owns 64 output columns; per step computes
//   h_new[0:64, cols] = tanh(xw[t] + h_prev @ W_hh[:, cols])
// W_hh column slice lives transposed in LDS for the whole kernel; prev-h
// tiles are double-buffered through LDS; h itself lives inside d_out.
// ---------------------------------------------------------------------------
__global__ __launch_bounds__(256) void birnn_scan(
    float* __restrict__ out,          // (32768, 1024) + hT tail
    const float* __restrict__ Whf,    // (512, 512)
    const float* __restrict__ Whb,    // (512, 512)
    unsigned* __restrict__ cnt)       // (2*T,) arrival counters, pre-zeroed
{
  extern __shared__ float smem[];
  float* __restrict__ Wt = smem;                  // 64 x 516, Wt[n][k]
  float* __restrict__ Hs = smem + 64 * WT_STRIDE; // 2 x 64 x 36

  const int tid  = threadIdx.x;
  const int wv   = tid >> 5;
  const int lane = tid & 31;
  const int l15  = lane & 15;
  const int hs   = lane >> 4;

  const int dir = blockIdx.x >> 3;    // 0=fwd, 1=bwd
  const int wg  = blockIdx.x & 7;
  const float* __restrict__ W = dir ? Whb : Whf;

  const int ncl    = (wv & 3) * 16;             // wave col within WG slice
  const int ncol   = wg * NSL + ncl;            // column base within NH
  const int mbase  = (wv >> 2) * 32;            // rows 0..31 or 32..63
  const int colOut = dir * NH + ncol;           // column within out row

  // ---- one-time: transpose this WG's W_hh column slice into LDS ----
  {
    const float* __restrict__ Wg = W + wg * NSL;
#pragma unroll 4
    for (int q = 0; q < 32; ++q) {
      const int idx = q * 256 + tid;            // 0..8191
      const int k  = idx >> 4;
      const int n4 = (idx & 15) * 4;
      const float4 v = *(const float4*)&Wg[(size_t)k * NH + n4];
      Wt[(n4 + 0) * WT_STRIDE + k] = v.x;
      Wt[(n4 + 1) * WT_STRIDE + k] = v.y;
      Wt[(n4 + 2) * WT_STRIDE + k] = v.z;
      Wt[(n4 + 3) * WT_STRIDE + k] = v.w;
    }
  }
  __syncthreads();

  // per-thread staging coords for a 64x32 h tile (2 float4 per thread)
  const int r0 = tid >> 3;            // 0..31
  const int r1 = r0 + 32;             // 32..63
  const int c0 = (tid & 7) * 4;

  for (int s = 0; s < TT; ++s) {
    const int t  = dir ? (TT - 1 - s) : s;
    const int tp = dir ? (t + 1) : (t - 1);

    // accumulators initialized from xw[t] (bias already folded in)
    v8f acc[2];
#pragma unroll
    for (int i = 0; i < 2; ++i)
#pragma unroll
      for (int r = 0; r < 8; ++r)
        acc[i][r] =
            out[(size_t)(t * BB + mbase + i * 16 + hs * 8 + r) * OUTW + colOut + l15];

    if (s > 0) {                       // h0 == 0 -> skip GEMM at s==0
      const size_t hbase = (size_t)tp * BB * OUTW + dir * NH;

      // prologue: stage k-tile 0 into buffer 0
#if HAVE_ASYNC
      async_cp16(&out[hbase + (size_t)r0 * OUTW + c0], &Hs[r0 * HS_STRIDE + c0]);
      async_cp16(&out[hbase + (size_t)r1 * OUTW + c0], &Hs[r1 * HS_STRIDE + c0]);
      asm volatile("s_wait_asynccnt 0x0" ::: "memory");
#else
      {
        const float4 v0 = *(const float4*)&out[hbase + (size_t)r0 * OUTW + c0];
        const float4 v1 = *(const float4*)&out[hbase + (size_t)r1 * OUTW + c0];
        *(float4*)&Hs[r0 * HS_STRIDE + c0] = v0;
        *(float4*)&Hs[r1 * HS_STRIDE + c0] = v1;
      }
#endif
      __syncthreads();

      for (int i = 0; i < 16; ++i) {
        const int kb = i * 32;
        const float* __restrict__ hb = Hs + (i & 1) * HS_BUF;
        float* __restrict__ hn = Hs + ((i + 1) & 1) * HS_BUF;
        const int kbn = kb + 32;

        // issue staging of the next k-tile into the other buffer
#if HAVE_ASYNC
        if (i + 1 < 16) {
          async_cp16(&out[hbase + (size_t)r0 * OUTW + kbn + c0], &hn[r0 * HS_STRIDE + c0]);
          async_cp16(&out[hbase + (size_t)r1 * OUTW + kbn + c0], &hn[r1 * HS_STRIDE + c0]);
        }
#else
        float4 v0 = {}, v1 = {};
        if (i + 1 < 16) {
          v0 = *(const float4*)&out[hbase + (size_t)r0 * OUTW + kbn + c0];
          v1 = *(const float4*)&out[hbase + (size_t)r1 * OUTW + kbn + c0];
        }
#endif

#pragma unroll
        for (int kk = 0; kk < 8; ++kk) {
          const int kc = kk * 4 + 2 * hs;
          const v2f a0 = *(const v2f*)&hb[(mbase + l15) * HS_STRIDE + kc];
          const v2f a1 = *(const v2f*)&hb[(mbase + 16 + l15) * HS_STRIDE + kc];
          const v2f b  = *(const v2f*)&Wt[(ncl + l15) * WT_STRIDE + kb + kc];
          acc[0] = wmma4(a0, b, acc[0]);
          acc[1] = wmma4(a1, b, acc[1]);
        }

#if HAVE_ASYNC
        asm volatile("s_wait_asynccnt 0x0" ::: "memory");
#else
        if (i + 1 < 16) {
          *(float4*)&hn[r0 * HS_STRIDE + c0] = v0;
          *(float4*)&hn[r1 * HS_STRIDE + c0] = v1;
        }
#endif
        __syncthreads();
      }
    }

    // tanh + in-place store of h into out[t]
#pragma unroll
    for (int i = 0; i < 2; ++i) {
#pragma unroll
      for (int r = 0; r < 8; ++r) acc[i][r] = tanhf(acc[i][r]);
#pragma unroll
      for (int r = 0; r < 8; ++r)
        out[(size_t)(t * BB + mbase + i * 16 + hs * 8 + r) * OUTW + colOut + l15] =
            acc[i][r];
    }

    if (s + 1 < TT) {
      // release my slice, arrive, wait for all 8 WGs of this direction
      __threadfence();
      __syncthreads();
      if (tid == 0) {
        atomicAdd(&cnt[dir * TT + s], 1u);
        while (__hip_atomic_load(&cnt[dir * TT + s], __ATOMIC_RELAXED,
                                 __HIP_MEMORY_SCOPE_AGENT) < (unsigned)NWG_DIR)
          __builtin_amdgcn_s_sleep(1);
      }
      __syncthreads();
      __threadfence();                 // acquire other WGs' h slices
    } else {
      // final hidden state: f_H then b_H appended after outputs
      float* __restrict__ hT =
          out + (size_t)TT * BB * OUTW + (size_t)dir * BB * NH;
#pragma unroll
      for (int i = 0; i < 2; ++i)
#pragma unroll
        for (int r = 0; r < 8; ++r)
          hT[(size_t)(mbase + i * 16 + hs * 8 + r) * NH + ncol + l15] = acc[i][r];
    }
  }
}

// ---------------------------------------------------------------------------
extern "C" void kernel_launch(void* const* d_in, const int* in_sizes, int n_in,
                              void* d_out, int out_size, void* d_ws, size_t ws_size,
                              hipStream_t stream) {
  const float* inputs = (const float*)d_in[0];
  const float* Wxh_f  = (const float*)d_in[1];
  const float* Whh_f  = (const float*)d_in[2];
  const float* bh_f   = (const float*)d_in[3];
  const float* Wxh_b  = (const float*)d_in[4];
  const float* Whh_b  = (const float*)d_in[5];
  const float* bh_b   = (const float*)d_in[6];
  float*    out = (float*)d_out;
  unsigned* cnt = (unsigned*)d_ws;

  birnn_init_counters<<<dim3(1), dim3(1024), 0, stream>>>(cnt);
  birnn_xw_gemm<<<dim3(MROWS / 128, 16), dim3(256), 0, stream>>>(
      inputs, Wxh_f, Wxh_b, bh_f, bh_b, out);
  birnn_scan<<<dim3(2 * NWG_DIR), dim3(256), SCAN_LDS_BYTES, stream>>>(
      out, Whh_f, Whh_b, cnt);
}